// BasicConvolutionBlock_4063039062845
// MI455X (gfx1250) — compile-verified
//
#include <hip/hip_runtime.h>

#define N_VOX 250000
#define C_IN  64
#define C_OUT 64
#define K_OFF 27
#define BN_EPS 1e-5f

typedef __attribute__((ext_vector_type(2))) float v2f;
typedef __attribute__((ext_vector_type(8))) float v8f;

// d_ws layout: [0..255] floats: sum(64), sumsq(64), scale(64), bias(64)
// then the repacked W.
#define STATS_BYTES 1024
#define WPACK_FLOAT4 (K_OFF * 4 * 8 * 32)     // 27648 float4 = 442368 B

// ---------------------------------------------------------------------------
// Kernel 1: zero the conv accumulator (d_out) and the BN sum/sumsq area
// ---------------------------------------------------------------------------
__global__ void zero_kernel(float* __restrict__ out, float* __restrict__ stats,
                            int total) {
  int idx = blockIdx.x * blockDim.x + threadIdx.x;
  if (idx < total) out[idx] = 0.0f;
  if (idx < 2 * C_OUT) stats[idx] = 0.0f;
}

// ---------------------------------------------------------------------------
// Kernel 1b: repack W into WMMA-lane order so the conv kernel's B operand is
// 8 coalesced b128 loads per n-tile instead of 32 strided b32 loads.
// wpack[(((k*4 + nt)*8 + sp)*32 + lane)] (float4): for s0=2sp, s1=2sp+1,
// {B(s0).x, B(s0).y, B(s1).x, B(s1).y} with
// B(s).x = W[k][4s   + 2*half][n0 + l16]
// B(s).y = W[k][4s+1 + 2*half][n0 + l16]   (half=lane>>4, l16=lane&15)
// ---------------------------------------------------------------------------
__global__ void repack_kernel(const float* __restrict__ W,
                              float4* __restrict__ wpack) {
  int tid = blockIdx.x * blockDim.x + threadIdx.x;
  if (tid >= WPACK_FLOAT4) return;
  const int lane = tid & 31;
  const int sp   = (tid >> 5) & 7;
  const int nt   = (tid >> 8) & 3;
  const int k    = tid >> 10;
  const int half = lane >> 4;
  const int col  = nt * 16 + (lane & 15);
  const int s0   = 2 * sp, s1 = 2 * sp + 1;
  const float* wk = W + (size_t)k * C_IN * C_OUT;
  float4 v;
  v.x = wk[(4 * s0 + 2 * half) * C_OUT + col];
  v.y = wk[(4 * s0 + 1 + 2 * half) * C_OUT + col];
  v.z = wk[(4 * s1 + 2 * half) * C_OUT + col];
  v.w = wk[(4 * s1 + 1 + 2 * half) * C_OUT + col];
  wpack[tid] = v;
}

// ---------------------------------------------------------------------------
// Kernel 2: gather -> WMMA fp32 GEMM -> atomic scatter-add
// One wave per 16-row tile per kernel offset k (grid.y = k so W[k] stays hot).
// ---------------------------------------------------------------------------
template <bool PACKED>
__global__ __launch_bounds__(256) void conv_kernel(
    const float*  __restrict__ feats,     // [N, 64]
    const float*  __restrict__ W,         // [27, 64, 64]
    const float4* __restrict__ wpack,     // repacked W (or nullptr)
    const int*    __restrict__ in_map,    // [27, N]
    const int*    __restrict__ out_map,   // [27, N]
    float*        __restrict__ acc)       // [N, 64]
{
  const int lane = threadIdx.x & 31;
  const int wave = threadIdx.x >> 5;
  const int k    = blockIdx.y;
  const int NT   = N_VOX / 16;            // 15625, exact
  const int tile = blockIdx.x * 8 + wave; // wave-uniform
  if (tile >= NT) return;                 // whole-wave exit: EXEC stays all-1s

  const int i0   = tile * 16;
  const int half = lane >> 4;
  const int l16  = lane & 15;

  // --- Gather A tile (16x64 fp32) in 16x16x4 fp32 A layout:
  // lane<16 holds K=4s,4s+1 of row l16; lane>=16 holds K=4s+2,4s+3.
  const long long arow = (long long)in_map[(long long)k * N_VOX + i0 + l16];
  const float* abase = feats + arow * C_IN + 2 * half;
  v2f a[16];
#pragma unroll
  for (int s = 0; s < 16; ++s)
    a[s] = *(const v2f*)(abase + 4 * s);

  // --- Scatter addresses: D layout => this lane covers M = r + 8*half,
  // N = l16, vgpr r = M&7. Load 8 contiguous ints as 2x int4 (16B aligned),
  // then form the 8 byte-addresses ONCE (sext + shift hoisted out of nt loop).
  const int4* omv =
      (const int4*)(out_map + (long long)k * N_VOX + i0 + 8 * half);
  const int4 o0 = omv[0], o1 = omv[1];
  const int omi[8] = {o0.x, o0.y, o0.z, o0.w, o1.x, o1.y, o1.z, o1.w};
  float* pr[8];
#pragma unroll
  for (int r = 0; r < 8; ++r)
    pr[r] = acc + (long long)omi[r] * C_OUT + l16;

  const float* wk = W + (size_t)k * C_IN * C_OUT;

#pragma unroll
  for (int nt = 0; nt < 4; ++nt) {        // 4 x 16 columns of C_OUT
    const int n0 = nt * 16;
    v8f c = {};
    if (PACKED) {
      const float4* wp = wpack + ((size_t)(k * 4 + nt) * 8) * 32 + lane;
#pragma unroll
      for (int sp = 0; sp < 8; ++sp) {    // 2 K-steps per b128
        const float4 t = wp[(size_t)sp * 32];
        v2f b0; b0.x = t.x; b0.y = t.y;
        v2f b1; b1.x = t.z; b1.y = t.w;
        c = __builtin_amdgcn_wmma_f32_16x16x4_f32(
            false, a[2 * sp], false, b0, (short)0, c, false, false);
        c = __builtin_amdgcn_wmma_f32_16x16x4_f32(
            false, a[2 * sp + 1], false, b1, (short)0, c, false, false);
      }
    } else {
      const float* wb = wk + (size_t)(2 * half) * C_OUT + (n0 + l16);
#pragma unroll
      for (int s = 0; s < 16; ++s) {
        v2f b;
        b.x = wb[(4 * s) * C_OUT];
        b.y = wb[(4 * s + 1) * C_OUT];
        c = __builtin_amdgcn_wmma_f32_16x16x4_f32(
            false, a[s], false, b, (short)0, c, false, false);
      }
    }
    // Scatter-accumulate the 16x16 D tile; accumulator is L2-resident
    // (64MB of 192MB), so fp32 atomics resolve on-chip.
#pragma unroll
    for (int r = 0; r < 8; ++r) {
      atomicAdd(pr[r] + n0, c[r]);
    }
  }
}

// ---------------------------------------------------------------------------
// Kernel 3: per-channel sum / sum-of-squares over all N voxels
// ---------------------------------------------------------------------------
__global__ __launch_bounds__(256) void stats_kernel(
    const float* __restrict__ acc, float* __restrict__ stats, int total) {
  __shared__ float ssum[256];
  __shared__ float ssq[256];
  const int tid = threadIdx.x;
  float s = 0.0f, q = 0.0f;
  for (long long idx = (long long)blockIdx.x * blockDim.x + tid; idx < total;
       idx += (long long)gridDim.x * blockDim.x) {
    float v = acc[idx];
    s += v;
    q += v * v;
  }
  ssum[tid] = s;
  ssq[tid]  = q;
  __syncthreads();
  if (tid < C_OUT) {
    float S = ssum[tid] + ssum[tid + 64] + ssum[tid + 128] + ssum[tid + 192];
    float Q = ssq[tid]  + ssq[tid + 64]  + ssq[tid + 128]  + ssq[tid + 192];
    atomicAdd(&stats[tid], S);
    atomicAdd(&stats[C_OUT + tid], Q);
  }
}

// ---------------------------------------------------------------------------
// Kernel 3b: fold BN stats + gamma/beta into per-channel scale/bias
// stats[128+c] = scale, stats[192+c] = bias
// ---------------------------------------------------------------------------
__global__ void finalize_kernel(float* __restrict__ stats,
                                const float* __restrict__ gamma,
                                const float* __restrict__ beta) {
  int c = threadIdx.x;
  if (c >= C_OUT) return;
  const float invN = 1.0f / (float)N_VOX;
  const float mean = stats[c] * invN;
  const float var  = stats[C_OUT + c] * invN - mean * mean;
  const float sc   = __frsqrt_rn(var + BN_EPS) * gamma[c];
  stats[2 * C_OUT + c] = sc;
  stats[3 * C_OUT + c] = beta[c] - mean * sc;
}

// ---------------------------------------------------------------------------
// Kernel 4: y = max(x*scale + bias, 0), in place, float4
// ---------------------------------------------------------------------------
__global__ __launch_bounds__(256) void norm_kernel(
    float* __restrict__ out, const float* __restrict__ stats, int total4) {
  long long idx = (long long)blockIdx.x * blockDim.x + threadIdx.x;
  if (idx >= total4) return;
  const int c0 = (int)((idx * 4) & (C_OUT - 1));
  const float4 sc = *(const float4*)(stats + 2 * C_OUT + c0);
  const float4 bs = *(const float4*)(stats + 3 * C_OUT + c0);
  float4 v = ((const float4*)out)[idx];
  v.x = fmaxf(fmaf(v.x, sc.x, bs.x), 0.0f);
  v.y = fmaxf(fmaf(v.y, sc.y, bs.y), 0.0f);
  v.z = fmaxf(fmaf(v.z, sc.z, bs.z), 0.0f);
  v.w = fmaxf(fmaf(v.w, sc.w, bs.w), 0.0f);
  ((float4*)out)[idx] = v;
}

// ---------------------------------------------------------------------------
extern "C" void kernel_launch(void* const* d_in, const int* in_sizes, int n_in,
                              void* d_out, int out_size, void* d_ws,
                              size_t ws_size, hipStream_t stream) {
  const float* feats   = (const float*)d_in[0];
  const float* W       = (const float*)d_in[1];
  const float* gamma   = (const float*)d_in[2];
  const float* beta    = (const float*)d_in[3];
  const int*   in_map  = (const int*)d_in[4];
  const int*   out_map = (const int*)d_in[5];
  float*  out   = (float*)d_out;
  float*  stats = (float*)d_ws;                              // 256 floats
  float4* wpack = (float4*)((char*)d_ws + STATS_BYTES);      // 442 KB

  const int total = N_VOX * C_OUT;  // 16,000,000
  const bool use_packed =
      ws_size >= (size_t)STATS_BYTES + (size_t)WPACK_FLOAT4 * sizeof(float4);

  zero_kernel<<<(total + 255) / 256, 256, 0, stream>>>(out, stats, total);

  dim3 cgrid((N_VOX / 16 + 7) / 8, K_OFF);  // 8 waves/block, one tile/wave
  if (use_packed) {
    repack_kernel<<<(WPACK_FLOAT4 + 255) / 256, 256, 0, stream>>>(W, wpack);
    conv_kernel<true><<<cgrid, 256, 0, stream>>>(feats, W, wpack, in_map,
                                                 out_map, out);
  } else {
    conv_kernel<false><<<cgrid, 256, 0, stream>>>(feats, W, wpack, in_map,
                                                  out_map, out);
  }

  stats_kernel<<<1024, 256, 0, stream>>>(out, stats, total);
  finalize_kernel<<<1, 64, 0, stream>>>(stats, gamma, beta);

  const int total4 = total / 4;
  norm_kernel<<<(total4 + 255) / 256, 256, 0, stream>>>(out, stats, total4);
}